// ChamferDistance2D_91139206021230
// MI455X (gfx1250) — compile-verified
//
#include <hip/hip_runtime.h>

typedef __attribute__((ext_vector_type(2))) float v2f;
typedef __attribute__((ext_vector_type(8))) float v8f;

#define BATCH 8
#define NPTS 8192
#define ROWS_PER_BLOCK 128   // 8 waves * 16 rows each
#define TILE_M 16

// Computes, for each "row" point, min over all "scanned" points of the squared
// distance, using V_WMMA_F32_16X16X4_F32 on the expanded K=4 form:
//   sqdist = [ax, ay, |a|^2, 1] . [-2bx, -2by, 1, |b|^2]
// Each block emits one partial sum (sum of row-mins for its 128 rows).
__global__ __launch_bounds__(256)
void chamfer_wmma_kernel(const float2* __restrict__ pts1,
                         const float2* __restrict__ pts2,
                         float* __restrict__ partials)
{
    __shared__ float2 spts[NPTS];   // 64KB: scanned point set for this batch

    const int lane = threadIdx.x & 31;
    const int wave = threadIdx.x >> 5;
    const int b    = blockIdx.y;
    const int dir  = blockIdx.z;

    const float2* __restrict__ P = (dir == 0) ? pts1 : pts2;  // row points
    const float2* __restrict__ Q = (dir == 0) ? pts2 : pts1;  // scanned points

    // Cooperative stage of Q[b] into LDS (L2-resident source, 64KB)
    for (int i = threadIdx.x; i < NPTS; i += 256)
        spts[i] = Q[b * NPTS + i];
    __syncthreads();

    // A fragment: 16x4 f32, 2 VGPRs/lane.
    // lanes 0-15 : row = lane,    v0 = K0 = ax,   v1 = K1 = ay
    // lanes 16-31: row = lane-16, v0 = K2 = |a|^2, v1 = K3 = 1
    const int  row = blockIdx.x * ROWS_PER_BLOCK + wave * 16 + (lane & 15);
    const bool lo  = (lane < 16);
    float2 p = P[b * NPTS + row];
    v2f afrag;
    afrag.x = lo ? p.x : (p.x * p.x + p.y * p.y);
    afrag.y = lo ? p.y : 1.0f;

    float mins[8];
#pragma unroll
    for (int r = 0; r < 8; ++r) mins[r] = 3.402823466e38f;

#pragma unroll 4
    for (int m0 = 0; m0 < NPTS; m0 += TILE_M) {
        // B fragment: 4x16 f32, lane = column.
        // lanes 0-15 : v0 = K0 = -2bx, v1 = K1 = -2by
        // lanes 16-31: v0 = K2 = 1,    v1 = K3 = |b|^2
        float2 q = spts[m0 + (lane & 15)];
        v2f bfrag;
        bfrag.x = lo ? (-2.0f * q.x) : 1.0f;
        bfrag.y = lo ? (-2.0f * q.y) : (q.x * q.x + q.y * q.y);

        v8f c = {};
        // D = A x B : 16x16 tile of squared distances, exact f32
        v8f d = __builtin_amdgcn_wmma_f32_16x16x4_f32(
            /*neg_a=*/false, afrag, /*neg_b=*/false, bfrag,
            /*c_mod=*/(short)0, c, /*reuse_a=*/false, /*reuse_b=*/false);

#pragma unroll
        for (int r = 0; r < 8; ++r) mins[r] = fminf(mins[r], d[r]);
    }

    // Cross-lane min over the 16 columns held by each half-wave.
    // D layout: VGPR r, lanes 0-15 -> row r, lanes 16-31 -> row r+8.
#pragma unroll
    for (int r = 0; r < 8; ++r) {
        float v = mins[r];
        v = fminf(v, __shfl_xor(v, 1, 32));
        v = fminf(v, __shfl_xor(v, 2, 32));
        v = fminf(v, __shfl_xor(v, 4, 32));
        v = fminf(v, __shfl_xor(v, 8, 32));
        mins[r] = v;
    }
    // lanes 0-15 now hold final mins of rows 0-7; lanes 16-31 rows 8-15
    float s = 0.0f;
#pragma unroll
    for (int r = 0; r < 8; ++r) s += mins[r];
    s += __shfl_xor(s, 16, 32);   // every lane: sum of this wave's 16 row-mins

    __syncthreads();              // all waves done reading spts; reuse as scratch
    float* wsums = (float*)spts;
    if (lane == 0) wsums[wave] = s;
    __syncthreads();

    if (threadIdx.x == 0) {
        float t = 0.0f;
#pragma unroll
        for (int w = 0; w < 8; ++w) t += wsums[w];
        const int pidx = (blockIdx.z * gridDim.y + blockIdx.y) * gridDim.x + blockIdx.x;
        partials[pidx] = t;
    }
}

// Deterministic fixed-order tree reduction of the 1024 block partials.
__global__ __launch_bounds__(1024)
void chamfer_reduce_kernel(const float* __restrict__ partials,
                           float* __restrict__ out, int n, float scale)
{
    __shared__ float buf[1024];
    const int t = threadIdx.x;
    buf[t] = (t < n) ? partials[t] : 0.0f;
    __syncthreads();
    for (int s = 512; s > 0; s >>= 1) {
        if (t < s) buf[t] += buf[t + s];
        __syncthreads();
    }
    if (t == 0) out[0] = buf[0] * scale;
}

extern "C" void kernel_launch(void* const* d_in, const int* in_sizes, int n_in,
                              void* d_out, int out_size, void* d_ws, size_t ws_size,
                              hipStream_t stream)
{
    (void)in_sizes; (void)n_in; (void)out_size; (void)ws_size;
    const float2* pts1 = (const float2*)d_in[0];   // [B, N, 2] f32
    const float2* pts2 = (const float2*)d_in[1];   // [B, M, 2] f32
    float* partials = (float*)d_ws;                // 1024 floats of scratch

    dim3 grid(NPTS / ROWS_PER_BLOCK, BATCH, 2);    // (64, 8, 2)
    chamfer_wmma_kernel<<<grid, 256, 0, stream>>>(pts1, pts2, partials);

    const int nPartials = grid.x * grid.y * grid.z;            // 1024
    chamfer_reduce_kernel<<<1, 1024, 0, stream>>>(partials, (float*)d_out,
                                                  nPartials, 1.0f / (float)NPTS);
}